// TranscriptomicModality_41120016892160
// MI455X (gfx1250) — compile-verified
//
#include <hip/hip_runtime.h>
#include <math.h>
#include <stdint.h>

// Problem constants (from reference)
constexpr int NN  = 100000;   // nodes (divisible by 16)
constexpr int EE  = 1000000;  // edges (divisible by 8)
constexpr int DIN = 256;
constexpr int HID = 64;
constexpr int OUTC = 32;

typedef float v2f __attribute__((ext_vector_type(2)));
typedef float v8f __attribute__((ext_vector_type(8)));

__device__ __forceinline__ float waveReduceSum(float v) {
#pragma unroll
  for (int off = 16; off >= 1; off >>= 1) v += __shfl_xor(v, off, 32);
  return v;
}

// ---------------------------------------------------------------------------
// Y[M, NT*16] = X[M,K] @ W[K, NT*16] + bias  via V_WMMA_F32_16X16X4_F32.
// One wave32 computes a full 16-row x Nout strip: NT independent 16x16
// accumulators. The A fragment (16x4 f32) is loaded ONCE per K-step and
// reused by all NT column tiles (12x less X traffic than tile-per-wave);
// B is small (<=192KB) and L2-resident. NT accumulators give independent
// WMMA chains that hide the WMMA->WMMA RAW latency.
// A layout: lanes 0-15 M=0..15; VGPR0 = K0(lo)/K2(hi), VGPR1 = K1/K3.
// B layout: mirrored (lanes = N, VGPRs = K).
// C/D: VGPR r -> M=r (lanes 0-15), M=8+r (lanes 16-31).
// ---------------------------------------------------------------------------
template <int NT>
__global__ __launch_bounds__(256) void gemm_strip_wmma(
    const float* __restrict__ X, const float* __restrict__ W,
    const float* __restrict__ bias, float* __restrict__ Y, int M, int K) {
  constexpr int Nout = NT * 16;
  const int lane = threadIdx.x & 31;
  const int wave = blockIdx.x * (blockDim.x >> 5) + (threadIdx.x >> 5);
  if (wave >= (M >> 4)) return;             // wave-uniform: EXEC stays all-ones
  const int row = wave << 4;
  const int mn  = lane & 15;                // A row / B col for this lane
  const int khi = (lane >> 4) << 1;         // 0 (lanes 0-15) or 2 (lanes 16-31)

  v8f acc[NT];
#pragma unroll
  for (int t = 0; t < NT; ++t) acc[t] = v8f{};

  const float* xrow  = X + (size_t)(row + mn) * K + khi;   // 8B-aligned
  const float* wbase = W + (size_t)khi * Nout + mn;
  for (int k = 0; k < K; k += 4) {
    const v2f a = *(const v2f*)(xrow + k);                 // global_load_b64
    const float* wk = wbase + (size_t)k * Nout;
#pragma unroll
    for (int t = 0; t < NT; ++t) {
      v2f b;
      b.x = wk[t * 16];
      b.y = wk[Nout + t * 16];
      acc[t] = __builtin_amdgcn_wmma_f32_16x16x4_f32(false, a, false, b,
                                                     (short)0, acc[t],
                                                     false, false);
    }
  }

  const int mb = (lane >> 4) << 3;          // 0 or 8
#pragma unroll
  for (int t = 0; t < NT; ++t) {
    const float bv = bias[t * 16 + mn];
    float* yout = Y + (size_t)(row + mb) * Nout + t * 16 + mn;
#pragma unroll
    for (int r = 0; r < 8; ++r) yout[(size_t)r * Nout] = acc[t][r] + bv;
  }
}

// ---------------------------------------------------------------------------
__global__ __launch_bounds__(256) void zero_kernel(float* __restrict__ p, size_t n) {
  size_t i = (size_t)blockIdx.x * blockDim.x + threadIdx.x;
  if (i < n) p[i] = 0.0f;
}

// Monotone uint encoding of float for integer atomicMax-based segment max
__device__ __forceinline__ unsigned encF(float v) {
  unsigned b = __float_as_uint(v);
  return (b & 0x80000000u) ? ~b : (b | 0x80000000u);
}
__device__ __forceinline__ float decF(unsigned e) {
  unsigned b = (e & 0x80000000u) ? (e & 0x7fffffffu) : ~e;
  return __uint_as_float(b);
}

// ---------------------------------------------------------------------------
// Pass 1: per-edge logit = dot(q[dst], k[src]) * scale; segment max over dst.
// One wave32 per edge; lanes stride channels (coalesced 64/32-float gathers).
// ---------------------------------------------------------------------------
__global__ __launch_bounds__(256) void edge_logits_max(
    const float* __restrict__ qkv, const int* __restrict__ src,
    const int* __restrict__ dst, float* __restrict__ logits,
    unsigned* __restrict__ maxenc, int nedges, int c, float scale) {
  const int lane = threadIdx.x & 31;
  const int e = blockIdx.x * (blockDim.x >> 5) + (threadIdx.x >> 5);
  if (e >= nedges) return;
  const int s = src[e], d = dst[e];
  const float* q = qkv + (size_t)d * (3 * c);       // query at dst
  const float* k = qkv + (size_t)s * (3 * c) + c;   // key at src
  float acc = 0.f;
  for (int i = lane; i < c; i += 32) acc += q[i] * k[i];
  acc = waveReduceSum(acc) * scale;
  if (lane == 0) {
    logits[e] = acc;
    __hip_atomic_fetch_max(maxenc + d, encF(acc),
                           __ATOMIC_RELAXED, __HIP_MEMORY_SCOPE_AGENT);
  }
}

// ---------------------------------------------------------------------------
// Pass 2: p = exp(logit - m[dst]); denom[dst] += p; agg[dst][:] += p * v[src][:]
// ---------------------------------------------------------------------------
__global__ __launch_bounds__(256) void edge_exp_scatter(
    const float* __restrict__ qkv, const int* __restrict__ src,
    const int* __restrict__ dst, const float* __restrict__ logits,
    const unsigned* __restrict__ maxenc, float* __restrict__ denom,
    float* __restrict__ agg, int nedges, int c) {
  const int lane = threadIdx.x & 31;
  const int e = blockIdx.x * (blockDim.x >> 5) + (threadIdx.x >> 5);
  if (e >= nedges) return;
  const int s = src[e], d = dst[e];
  const unsigned enc = maxenc[d];
  const float m = (enc == 0u) ? 0.0f : decF(enc);
  const float p = __expf(logits[e] - m);
  if (lane == 0)
    __hip_atomic_fetch_add(denom + d, p, __ATOMIC_RELAXED, __HIP_MEMORY_SCOPE_AGENT);
  const float* v = qkv + (size_t)s * (3 * c) + 2 * c;
  float* ag = agg + (size_t)d * c;
  for (int i = lane; i < c; i += 32)
    __hip_atomic_fetch_add(ag + i, p * v[i], __ATOMIC_RELAXED, __HIP_MEMORY_SCOPE_AGENT);
}

// ---------------------------------------------------------------------------
// Finalize (layers 1-3, c == 64): h = agg/denom + skip; LayerNorm; ReLU.
// One wave32 per node; each lane owns channels {lane, lane+32}.
// ---------------------------------------------------------------------------
__global__ __launch_bounds__(256) void finalize_ln(
    const float* __restrict__ agg, const float* __restrict__ denom,
    const float* __restrict__ skip, const float* __restrict__ g,
    const float* __restrict__ b, float* __restrict__ h, int n) {
  const int lane = threadIdx.x & 31;
  const int node = blockIdx.x * (blockDim.x >> 5) + (threadIdx.x >> 5);
  if (node >= n) return;
  const float dn = denom[node] + 1e-16f;
  const size_t base = (size_t)node * 64;
  float x0 = agg[base + lane]      / dn + skip[base + lane];
  float x1 = agg[base + lane + 32] / dn + skip[base + lane + 32];
  const float mean = waveReduceSum(x0 + x1) * (1.0f / 64.0f);
  const float d0 = x0 - mean, d1 = x1 - mean;
  const float var = waveReduceSum(d0 * d0 + d1 * d1) * (1.0f / 64.0f);
  const float rs = rsqrtf(var + 1e-5f);
  h[base + lane]      = fmaxf(d0 * rs * g[lane]      + b[lane],      0.0f);
  h[base + lane + 32] = fmaxf(d1 * rs * g[lane + 32] + b[lane + 32], 0.0f);
}

// ---------------------------------------------------------------------------
// Finalize (layer 4, c == 32): h = relu(agg/denom + skip); L2 normalize.
// One wave32 per node; lane == channel.
// ---------------------------------------------------------------------------
__global__ __launch_bounds__(256) void finalize_out(
    const float* __restrict__ agg, const float* __restrict__ denom,
    const float* __restrict__ skip, float* __restrict__ out, int n) {
  const int lane = threadIdx.x & 31;
  const int node = blockIdx.x * (blockDim.x >> 5) + (threadIdx.x >> 5);
  if (node >= n) return;
  const float dn = denom[node] + 1e-16f;
  const size_t base = (size_t)node * 32;
  float x = fmaxf(agg[base + lane] / dn + skip[base + lane], 0.0f);
  const float ss = waveReduceSum(x * x);
  const float nrm = fmaxf(sqrtf(ss), 1e-12f);
  out[base + lane] = x / nrm;
}

// ---------------------------------------------------------------------------
extern "C" void kernel_launch(void* const* d_in, const int* in_sizes, int n_in,
                              void* d_out, int out_size, void* d_ws, size_t ws_size,
                              hipStream_t stream) {
  const float* x   = (const float*)d_in[0];
  const int*   src = (const int*)d_in[1];
  const int*   dst = src + EE;

  const float *wq[4], *bq[4], *ws_[4], *bs[4];
  for (int i = 0; i < 4; ++i) {
    wq[i]  = (const float*)d_in[2 + 4 * i + 0];
    bq[i]  = (const float*)d_in[2 + 4 * i + 1];
    ws_[i] = (const float*)d_in[2 + 4 * i + 2];
    bs[i]  = (const float*)d_in[2 + 4 * i + 3];
  }
  const float* lng[3] = {(const float*)d_in[18], (const float*)d_in[20], (const float*)d_in[22]};
  const float* lnb[3] = {(const float*)d_in[19], (const float*)d_in[21], (const float*)d_in[23]};

  // Workspace layout (floats): qkv | skip | h | logits | agg | denom | maxenc
  char* p = (char*)d_ws;
  float*    qkvb   = (float*)p;    p += (size_t)NN * 192 * sizeof(float);
  float*    skipb  = (float*)p;    p += (size_t)NN * 64  * sizeof(float);
  float*    hbuf   = (float*)p;    p += (size_t)NN * 64  * sizeof(float);
  float*    logb   = (float*)p;    p += (size_t)EE       * sizeof(float);
  float*    aggb   = (float*)p;    p += (size_t)NN * 64  * sizeof(float);
  float*    denomb = (float*)p;    p += (size_t)NN       * sizeof(float);
  unsigned* maxenc = (unsigned*)p; // N unsigned

  const int WPB = 8;                    // waves per 256-thread block (wave32)
  const int gemmBlocks = ((NN / 16) + WPB - 1) / WPB;   // one wave per 16-row strip

  auto run_edges = [&](int c, int li, float* hout, bool last) {
    const size_t zn = (size_t)NN * 66;  // agg(64N) + denom(N) + maxenc(N), contiguous
    zero_kernel<<<(unsigned)((zn + 255) / 256), 256, 0, stream>>>(aggb, zn);
    const float scale = 1.0f / sqrtf((float)c);
    edge_logits_max<<<EE / WPB, 256, 0, stream>>>(qkvb, src, dst, logb, maxenc, EE, c, scale);
    edge_exp_scatter<<<EE / WPB, 256, 0, stream>>>(qkvb, src, dst, logb, maxenc, denomb, aggb, EE, c);
    if (!last) {
      finalize_ln<<<(NN + WPB - 1) / WPB, 256, 0, stream>>>(aggb, denomb, skipb,
                                                            lng[li], lnb[li], hout, NN);
    } else {
      finalize_out<<<(NN + WPB - 1) / WPB, 256, 0, stream>>>(aggb, denomb, skipb,
                                                             (float*)d_out, NN);
    }
  };

  // Layer 1: 256 -> 64
  gemm_strip_wmma<12><<<gemmBlocks, 256, 0, stream>>>(x, wq[0], bq[0], qkvb, NN, DIN);
  gemm_strip_wmma<4> <<<gemmBlocks, 256, 0, stream>>>(x, ws_[0], bs[0], skipb, NN, DIN);
  run_edges(HID, 0, hbuf, false);
  // Layer 2: 64 -> 64
  gemm_strip_wmma<12><<<gemmBlocks, 256, 0, stream>>>(hbuf, wq[1], bq[1], qkvb, NN, HID);
  gemm_strip_wmma<4> <<<gemmBlocks, 256, 0, stream>>>(hbuf, ws_[1], bs[1], skipb, NN, HID);
  run_edges(HID, 1, hbuf, false);
  // Layer 3: 64 -> 64
  gemm_strip_wmma<12><<<gemmBlocks, 256, 0, stream>>>(hbuf, wq[2], bq[2], qkvb, NN, HID);
  gemm_strip_wmma<4> <<<gemmBlocks, 256, 0, stream>>>(hbuf, ws_[2], bs[2], skipb, NN, HID);
  run_edges(HID, 2, hbuf, false);
  // Layer 4: 64 -> 32 (heads=1, concat=False == identity mean)
  gemm_strip_wmma<6> <<<gemmBlocks, 256, 0, stream>>>(hbuf, wq[3], bq[3], qkvb, NN, HID);
  gemm_strip_wmma<2> <<<gemmBlocks, 256, 0, stream>>>(hbuf, ws_[3], bs[3], skipb, NN, HID);
  run_edges(OUTC, 3, nullptr, true);
}